// StaticGeneralGraphConv_GGA_89481348645420
// MI455X (gfx1250) — compile-verified
//
#include <hip/hip_runtime.h>
#include <hip/hip_bf16.h>
#include <math.h>

// f32 WMMA fragment types for V_WMMA_F32_16X16X4_F32 (wave32):
//  A: 16x4 f32   -> 64 vals / 32 lanes = 2 VGPRs
//  C/D: 16x16 f32 -> 256 vals / 32 lanes = 8 VGPRs
typedef float v2f __attribute__((ext_vector_type(2)));
typedef float v8f __attribute__((ext_vector_type(8)));

// ---------------------------------------------------------------------------
// Elementwise helpers
// ---------------------------------------------------------------------------
__global__ void zero_f32(float* __restrict__ p, int n) {
    int i = blockIdx.x * blockDim.x + threadIdx.x;
    if (i < n) p[i] = 0.0f;
}

// e_norm = e / (sum(e^2) + 1e-8), streamed once (HBM-rate bound)
__global__ void normalize_e(const float* __restrict__ e, float* __restrict__ eout, int nE) {
    int i = blockIdx.x * blockDim.x + threadIdx.x;
    if (i < nE) {
        float a = e[3 * i + 0], b = e[3 * i + 1], c = e[3 * i + 2];
        float inv = 1.0f / (a * a + b * b + c * c + 1e-8f);
        eout[3 * i + 0] = a * inv;
        eout[3 * i + 1] = b * inv;
        eout[3 * i + 2] = c * inv;
    }
}

// ---------------------------------------------------------------------------
// One graph-conv layer: per edge  h = elu([x[dst], x[src], e] @ W1 + b1)
//                                 o = h @ W2 + b2 ;  atomic scatter-add at dst
// 16 edges per wave mapped onto V_WMMA_F32_16X16X4_F32 (M = 16 edges, K
// chunked by 4 over the concat dim, N = output channels zero-padded to 16).
// Bias folded into the C accumulator. Staging is full-wave:
//   - dst idx in lanes 0-15, src idx in lanes 16-31 (coalesced b32)
//   - e rows: 48 contiguous floats strided over all 32 lanes (coalesced)
//   - x gathers: one (side, edge) pair per lane, b128 row loads -> b128 LDS
// ---------------------------------------------------------------------------
template <int C_IN, int C_MID, int C_OUT>
__global__ __launch_bounds__(128) void edge_conv_wmma(
    const float* __restrict__ x, const int* __restrict__ srcIdx,
    const int* __restrict__ dstIdx, const float* __restrict__ e,
    const float* __restrict__ W1, const float* __restrict__ b1,
    const float* __restrict__ W2, const float* __restrict__ b2,
    float* __restrict__ xout, int nE)
{
    constexpr int K_CAT     = 2 * C_IN + 3;        // concat feature dim (11 or 19)
    constexpr int K1_CHUNKS = (K_CAT + 3) / 4;     // K=4 WMMA steps for GEMM1
    constexpr int K_PAD     = K1_CHUNKS * 4;       // zero-padded feature dim
    constexpr int K2_CHUNKS = (C_MID + 3) / 4;     // K=4 WMMA steps for GEMM2
    constexpr int WAVES     = 4;                   // 128 threads = 4 wave32
    constexpr int VEC       = C_IN / 4;            // float4's per node row (1 or 2)

    __shared__ alignas(16) float feat[WAVES][16][K_PAD];         // A for GEMM1
    __shared__ alignas(16) float hmid[WAVES][16][K2_CHUNKS * 4]; // D1^T -> A2
    __shared__ int dnode[WAVES][16];
    __shared__ int snode[WAVES][16];

    const int lane = threadIdx.x & 31;
    const int wv   = threadIdx.x >> 5;
    const int tile = blockIdx.x * WAVES + wv;
    const int e0   = tile * 16;

    // ---- Phase A1: coalesced index + e-row staging (all 32 lanes) ----
    {
        int m  = lane & 15;
        int ei = e0 + m;
        bool ok = ei < nE;
        if (lane < 16) dnode[wv][m] = ok ? dstIdx[ei] : -1;
        else           snode[wv][m] = ok ? srcIdx[ei] : -1;

        // 16 edges x 3 e-floats = 48 contiguous floats starting at e0*3
#pragma unroll
        for (int j = lane; j < 48; j += 32) {
            int f = e0 * 3 + j;                       // f < 3*nE implies edge ok
            feat[wv][j / 3][2 * C_IN + (j % 3)] = (f < 3 * nE) ? e[f] : 0.0f;
        }
        // zero-pad columns K_CAT..K_PAD-1
        if (lane < 16) {
#pragma unroll
            for (int k = K_CAT; k < K_PAD; ++k) feat[wv][m][k] = 0.0f;
        }
    }
    __syncthreads();

    // ---- Phase A2: x gathers, one (side, edge) per lane, b128 rows ----
    {
        int  m     = lane & 15;
        bool isSrc = lane >= 16;
        int  node  = isSrc ? snode[wv][m] : dnode[wv][m];
        int  base  = isSrc ? C_IN : 0;     // x[dst] first, then x[src]
        if (node >= 0) {
#pragma unroll
            for (int v = 0; v < VEC; ++v) {
                float4 r = ((const float4*)x)[node * VEC + v];
                *(float4*)&feat[wv][m][base + 4 * v] = r;   // 16B-aligned LDS
            }
        } else {
#pragma unroll
            for (int c = 0; c < C_IN; ++c) feat[wv][m][base + c] = 0.0f;
        }
    }
    __syncthreads();

    const int n  = lane & 15;   // N column for B/C/D; also M row for A fragments
    const int hf = lane >> 4;   // half-wave selects K (A/B) or M+8 (C/D)

    // ---- GEMM1: [16 x K_CAT] @ W1[K_CAT x C_MID] + b1 ----
    v8f acc;
#pragma unroll
    for (int r = 0; r < 8; ++r) acc[r] = (n < C_MID) ? b1[n] : 0.0f;
#pragma unroll
    for (int ch = 0; ch < K1_CHUNKS; ++ch) {
        const int kb = ch * 4 + 2 * hf;
        v2f a, b;
        a.x = feat[wv][n][kb];          // A: M=lane&15, K striped VGPR/half
        a.y = feat[wv][n][kb + 1];
        b.x = (kb     < K_CAT && n < C_MID) ? W1[(kb    ) * C_MID + n] : 0.0f;
        b.y = (kb + 1 < K_CAT && n < C_MID) ? W1[(kb + 1) * C_MID + n] : 0.0f;
        acc = __builtin_amdgcn_wmma_f32_16x16x4_f32(
            false, a, false, b, (short)0, acc, false, false);
    }

    // ---- ELU, transpose D1 (M in VGPR idx, N in lanes) through LDS ----
#pragma unroll
    for (int r = 0; r < 8; ++r) {
        int   m = r + 8 * hf;           // edge row of this D element
        float v = acc[r];
        v = v > 0.0f ? v : expm1f(v);   // elu
        if (n < C_MID) hmid[wv][m][n] = v;
    }
    __syncthreads();

    // ---- GEMM2: [16 x C_MID] @ W2[C_MID x C_OUT] + b2 ----
    v8f acc2;
#pragma unroll
    for (int r = 0; r < 8; ++r) acc2[r] = (n < C_OUT) ? b2[n] : 0.0f;
#pragma unroll
    for (int ch = 0; ch < K2_CHUNKS; ++ch) {
        const int kb = ch * 4 + 2 * hf;   // kb+1 < C_MID always (C_MID % 4 == 0)
        v2f a, b;
        a.x = hmid[wv][n][kb];
        a.y = hmid[wv][n][kb + 1];
        b.x = (n < C_OUT) ? W2[(kb    ) * C_OUT + n] : 0.0f;
        b.y = (n < C_OUT) ? W2[(kb + 1) * C_OUT + n] : 0.0f;
        acc2 = __builtin_amdgcn_wmma_f32_16x16x4_f32(
            false, a, false, b, (short)0, acc2, false, false);
    }

    // ---- Scatter-add to destination nodes (L2-resident accumulators) ----
#pragma unroll
    for (int r = 0; r < 8; ++r) {
        int m = r + 8 * hf;             // each lane owns a unique (edge, chan)
        int d = dnode[wv][m];
        if (d >= 0 && n < C_OUT)
            atomicAdd(&xout[d * C_OUT + n], acc2[r]);
    }
}

// ---------------------------------------------------------------------------
// Launch: inputs in setup_inputs() order.
// d_out = [x_final (100000 f32)] ++ [e_norm (3.2M x 3 f32)]
// d_ws  = x1 [N,8] ++ x2 [N,8]  (6.4 MB)
// ---------------------------------------------------------------------------
extern "C" void kernel_launch(void* const* d_in, const int* in_sizes, int n_in,
                              void* d_out, int out_size, void* d_ws, size_t ws_size,
                              hipStream_t stream) {
    const float* x    = (const float*)d_in[0];
    const int*   eidx = (const int*)d_in[1];
    const float* e    = (const float*)d_in[2];
    const float* W1a = (const float*)d_in[3];  const float* b1a = (const float*)d_in[4];
    const float* W2a = (const float*)d_in[5];  const float* b2a = (const float*)d_in[6];
    const float* W1b = (const float*)d_in[7];  const float* b1b = (const float*)d_in[8];
    const float* W2b = (const float*)d_in[9];  const float* b2b = (const float*)d_in[10];
    const float* W1c = (const float*)d_in[11]; const float* b1c = (const float*)d_in[12];
    const float* W2c = (const float*)d_in[13]; const float* b2c = (const float*)d_in[14];

    const int nE = in_sizes[2] / 3;   // 3,200,000
    const int nN = in_sizes[0] / 4;   // 100,000
    const int* srcI = eidx;           // edge_index[0]
    const int* dstI = eidx + nE;      // edge_index[1]

    float* out   = (float*)d_out;
    float* xfin  = out;               // [nN, 1]
    float* enorm = out + nN;          // [nE, 3] — also consumed by all 3 convs
    float* x1    = (float*)d_ws;                    // [nN, 8]
    float* x2    = x1 + (size_t)nN * 8;             // [nN, 8]

    normalize_e<<<(nE + 255) / 256, 256, 0, stream>>>(e, enorm, nE);

    const int tiles  = (nE + 15) / 16;
    const int blocks = (tiles + 3) / 4;   // 4 waves per block, 16 edges per wave

    zero_f32<<<(nN * 8 + 255) / 256, 256, 0, stream>>>(x1, nN * 8);
    edge_conv_wmma<4, 4, 8><<<blocks, 128, 0, stream>>>(
        x, srcI, dstI, enorm, W1a, b1a, W2a, b2a, x1, nE);

    zero_f32<<<(nN * 8 + 255) / 256, 256, 0, stream>>>(x2, nN * 8);
    edge_conv_wmma<8, 8, 8><<<blocks, 128, 0, stream>>>(
        x1, srcI, dstI, enorm, W1b, b1b, W2b, b2b, x2, nE);

    zero_f32<<<(nN + 255) / 256, 256, 0, stream>>>(xfin, nN);
    edge_conv_wmma<8, 4, 1><<<blocks, 128, 0, stream>>>(
        x2, srcI, dstI, enorm, W1c, b1c, W2c, b2c, xfin, nE);
}